// AttnBlock_71983651881459
// MI455X (gfx1250) — compile-verified
//
#include <hip/hip_runtime.h>
#include <hip/hip_bf16.h>

// Problem dims (fixed by the reference)
#define BB     4
#define HH     64
#define WW     64
#define DIMC   512
#define HEADS  64
#define QKV    8
#define NPOS   (BB*HH*WW)        // 16384 positions
#define SCALE  0.35355339059327373f   // 8^-0.5

typedef float v2f __attribute__((ext_vector_type(2)));
typedef float v8f __attribute__((ext_vector_type(8)));
typedef unsigned int v4u __attribute__((ext_vector_type(4)));
typedef int v4i __attribute__((ext_vector_type(4)));
typedef int v8i __attribute__((ext_vector_type(8)));

// ---------------------------------------------------------------------------
// TDM helper: issue a 2-D global->LDS tile copy on the Tensor Data Mover.
// D# built per CDNA5 ISA (CDNA5 ISA):
//   group0: [1:0]=count=1 | [63:32]=lds_addr | [120:64]=global_addr | [127:126]=type=2
//   group1: [17:16]=data_size(2 -> 4B) | tensor_dim0/1 | tile_dim0/1 | dim0_stride
// Groups 2/3 zero (tensor <= 2D). Tracked by TENSORcnt.
// dims/strides in 4-byte elements. 6-arg clang-23 builtin form.
// ---------------------------------------------------------------------------
__device__ __forceinline__ void tdm_load_2d_f32(void* lds_dst, const void* gsrc,
                                                unsigned tensor_d0, unsigned tensor_d1,
                                                unsigned tile_d0, unsigned tile_d1,
                                                unsigned d0_stride) {
  unsigned long long ga = (unsigned long long)(uintptr_t)gsrc;
  unsigned lds_addr = (unsigned)(uintptr_t)lds_dst;   // low 32b of flat = LDS byte addr
  v4u g0;
  g0[0] = 1u;                                               // count=1, user descriptor
  g0[1] = lds_addr;                                         // lds_addr (bytes)
  g0[2] = (unsigned)ga;                                     // global_addr[31:0]
  g0[3] = (unsigned)((ga >> 32) & 0x01FFFFFFu) | (2u << 30);// global_addr[56:32], type=2
  v8i g1;
  g1[0] = (int)(2u << 16);                                  // wg_mask=0, data_size=4B
  g1[1] = (int)((tensor_d0 & 0xFFFFu) << 16);               // tensor_dim0[15:0]
  g1[2] = (int)(((tensor_d0 >> 16) & 0xFFFFu) |
                ((tensor_d1 & 0xFFFFu) << 16));             // td0[31:16] | td1[15:0]
  g1[3] = (int)(((tensor_d1 >> 16) & 0xFFFFu) |
                ((tile_d0 & 0xFFFFu) << 16));               // td1[31:16] | tile_dim0
  g1[4] = (int)(tile_d1 & 0xFFFFu);                         // tile_dim1, tile_dim2=0
  g1[5] = (int)d0_stride;                                   // tensor_dim0_stride[31:0]
  g1[6] = 0;                                                // stride0[47:32], stride1[15:0]
  g1[7] = 0;
  v4i z4 = {0, 0, 0, 0};
  v8i z8 = {0, 0, 0, 0, 0, 0, 0, 0};
  __builtin_amdgcn_tensor_load_to_lds(g0, g1, z4, z4, z8, 0);  // cpol = 0
}

// ---------------------------------------------------------------------------
// Kernel 1: per-channel weight sums  wq_sum[c] = sum_{n,p} w_q[c,n,p]
// ---------------------------------------------------------------------------
__global__ void wsum_kernel(const float* __restrict__ wq,
                            const float* __restrict__ wk,
                            float* __restrict__ wqs,
                            float* __restrict__ wks) {
  const int lane = threadIdx.x & 31;
  const int wave = threadIdx.x >> 5;
  const int c = blockIdx.x * 8 + wave;            // 0..511
  const float* rq = wq + (size_t)c * (HEADS * QKV);
  const float* rk = wk + (size_t)c * (HEADS * QKV);
  float sq = 0.f, sk = 0.f;
  for (int i = lane; i < HEADS * QKV; i += 32) { sq += rq[i]; sk += rk[i]; }
  for (int o = 16; o; o >>= 1) {
    sq += __shfl_xor(sq, o, 32);
    sk += __shfl_xor(sk, o, 32);
  }
  if (lane == 0) { wqs[c] = sq; wks[c] = sk; }
}

// ---------------------------------------------------------------------------
// Kernel 2: qs[i] = x_row(i)·wq_sum ; ksS[i] = SCALE * x_row(i)·wk_sum
// ---------------------------------------------------------------------------
__global__ void qkdot_kernel(const float* __restrict__ x,
                             const float* __restrict__ wqs,
                             const float* __restrict__ wks,
                             float* __restrict__ qs,
                             float* __restrict__ ksS) {
  const int lane = threadIdx.x & 31;
  const int wave = threadIdx.x >> 5;
  const int i = blockIdx.x * 8 + wave;            // 0..16383
  const float* xr = x + (size_t)i * DIMC;
  float sq = 0.f, sk = 0.f;
  for (int c = lane; c < DIMC; c += 32) {
    float xv = xr[c];
    sq += xv * wqs[c];
    sk += xv * wks[c];
  }
  for (int o = 16; o; o >>= 1) {
    sq += __shfl_xor(sq, o, 32);
    sk += __shfl_xor(sk, o, 32);
  }
  if (lane == 0) { qs[i] = sq; ksS[i] = sk * SCALE; }
}

// ---------------------------------------------------------------------------
// Kernel 3: per (b,p) row stats of ksS (single-pass stable softmax with a
// scalar multiplier: rowmax(s*a) = s>=0 ? s*max(a) : s*min(a)).
// ---------------------------------------------------------------------------
__global__ void kstat_kernel(const float* __restrict__ ksS,
                             float* __restrict__ kmax,
                             float* __restrict__ kmin) {
  const int lane = threadIdx.x & 31;
  const int wave = threadIdx.x >> 5;
  const int r = blockIdx.x * 8 + wave;            // 0..255
  const float* row = ksS + (size_t)r * WW;
  float a = row[lane], b = row[lane + 32];
  float mx = fmaxf(a, b), mn = fminf(a, b);
  for (int o = 16; o; o >>= 1) {
    mx = fmaxf(mx, __shfl_xor(mx, o, 32));
    mn = fminf(mn, __shfl_xor(mn, o, 32));
  }
  if (lane == 0) { kmax[r] = mx; kmin[r] = mn; }
}

// ---------------------------------------------------------------------------
// Kernel 4: V projection GEMM  V[16384x512] = X[16384x512] @ Wv[512x512]
// fp32 WMMA 16x16x4, one wave per 16x16 C tile. The shared 16x512 A tile is
// DMA'd into LDS by the TDM (wave 0 issues, waits TENSORcnt, barrier), so
// all waves start their Wv loads only after the tile is in LDS.
//
// f32 fragment layouts (wave32):
//   A 16x4 : a[i], lane L -> M = L&15, K = 2*(L>>4)+i
//   B  4x16: b[i], lane L -> N = L&15, K = 2*(L>>4)+i
//   C 16x16: c[j], lane L -> N = L&15, M = j + 8*(L>>4)
// ---------------------------------------------------------------------------
__global__ void vproj_kernel(const float* __restrict__ x,
                             const float* __restrict__ wv,
                             float* __restrict__ v) {
  __shared__ __align__(16) float lds_a[16 * DIMC];   // 32 KB
  const int lane  = threadIdx.x & 31;
  const int wave  = threadIdx.x >> 5;
  const int mtile = blockIdx.x >> 2;                          // 0..1023
  const int ntile = ((blockIdx.x & 3) << 3) + wave;           // 0..31

  // TDM: stage A tile (16 rows x 512 f32) into LDS
  if (wave == 0) {
    tdm_load_2d_f32(lds_a, x + (size_t)mtile * 16 * DIMC,
                    /*tensor_d0=*/DIMC, /*tensor_d1=*/16,
                    /*tile_d0=*/DIMC, /*tile_d1=*/16,
                    /*d0_stride=*/DIMC);
    __builtin_amdgcn_s_wait_tensorcnt(0);
  }
  __syncthreads();

  const int n   = lane & 15;
  const int g2  = (lane >> 4) << 1;     // 0 or 2
  const int row = lane & 15;
  const int nn  = ntile * 16 + n;

  v8f c = {};
  for (int k0 = 0; k0 < DIMC; k0 += 4) {
    v2f a = *(const v2f*)&lds_a[row * DIMC + k0 + g2];        // ds_load_b64
    v2f b;
    b.x = wv[(size_t)(k0 + g2)     * (HEADS * QKV) + nn];
    b.y = wv[(size_t)(k0 + g2 + 1) * (HEADS * QKV) + nn];
    c = __builtin_amdgcn_wmma_f32_16x16x4_f32(false, a, false, b,
                                              (short)0, c, false, false);
  }

  float* dst = v + (size_t)mtile * 16 * (HEADS * QKV) + (size_t)ntile * 16 + n;
  const int mb = (lane >> 4) << 3;
#pragma unroll
  for (int j = 0; j < 8; ++j)
    dst[(size_t)(mb + j) * (HEADS * QKV)] = c[j];
}

// ---------------------------------------------------------------------------
// Kernel 5: attention epilogue. One wave per position idx=(b,h,w):
//   E[p,q] = exp(s*ksS[b,p,q] - m_p),  attn[p,d] = (E @ Vtile)[p,d] / rowsum
// E@Vtile via fp32 WMMA; B fragment holds Vtile in N=0..7 and ones at N=8 so
// C[p,8] = rowsum(E) falls out of the same WMMA. ksS[b] staged via TDM.
// ---------------------------------------------------------------------------
__global__ void attn_kernel(const float* __restrict__ qs,
                            const float* __restrict__ ksS,
                            const float* __restrict__ kmax,
                            const float* __restrict__ kmin,
                            const float* __restrict__ v,
                            float* __restrict__ out) {
  __shared__ __align__(16) float lds_k[HH * WW];     // 16 KB: ksS[b]
  const int lane = threadIdx.x & 31;
  const int wave = threadIdx.x >> 5;
  const int idx  = blockIdx.x * 8 + wave;            // 0..16383
  const int b    = idx >> 12;                        // 4096 positions per b

  if (wave == 0) {
    tdm_load_2d_f32(lds_k, ksS + (size_t)b * (HH * WW),
                    /*tensor_d0=*/HH * WW, /*tensor_d1=*/1,
                    /*tile_d0=*/HH * WW, /*tile_d1=*/1,
                    /*d0_stride=*/HH * WW);
    __builtin_amdgcn_s_wait_tensorcnt(0);
  }
  __syncthreads();

  const float s  = qs[idx];
  const int   n  = lane & 15;
  const int   g2 = (lane >> 4) << 1;

  // Preload B fragments: Vtile (64x8) in cols 0..7, ones in col 8, zeros 9..15
  const float* vp = v + (size_t)idx * (HEADS * QKV);
  v2f bb[16];
#pragma unroll
  for (int ks = 0; ks < 16; ++ks) {
    const int q0 = ks * 4 + g2;
#pragma unroll
    for (int i = 0; i < 2; ++i) {
      const int q = q0 + i;
      float val;
      if (n < 8)       val = vp[q * QKV + n];
      else if (n == 8) val = 1.0f;
      else             val = 0.0f;
      bb[ks][i] = val;
    }
  }

  // 4 M-tiles of 16 rows each
  for (int mt = 0; mt < 4; ++mt) {
    const int p = mt * 16 + n;                       // E-row handled by lane
    const float mrow = (s >= 0.f) ? s * kmax[b * 64 + p]
                                  : s * kmin[b * 64 + p];
    const float* krow = &lds_k[p * WW];
    v8f c = {};
#pragma unroll
    for (int ks = 0; ks < 16; ++ks) {
      v2f kk = *(const v2f*)&krow[ks * 4 + g2];      // ds_load_b64
      v2f a;
      a.x = __expf(s * kk.x - mrow);
      a.y = __expf(s * kk.y - mrow);
      c = __builtin_amdgcn_wmma_f32_16x16x4_f32(false, a, false, bb[ks],
                                                (short)0, c, false, false);
    }
    // normalize by rowsum (column N=8 of the same row) and store
    float* op = out + (size_t)idx * (HEADS * QKV) + (size_t)(mt * 16) * QKV;
    const int mb   = (lane >> 4) << 3;
    const int srcl = (lane & 16) | 8;                // lane holding N=8, same M-half
#pragma unroll
    for (int j = 0; j < 8; ++j) {
      float r = __shfl(c[j], srcl, 32);
      if (n < 8) op[(mb + j) * QKV + n] = c[j] / r;
    }
  }
}

// ---------------------------------------------------------------------------
// Launch
// ---------------------------------------------------------------------------
extern "C" void kernel_launch(void* const* d_in, const int* in_sizes, int n_in,
                              void* d_out, int out_size, void* d_ws, size_t ws_size,
                              hipStream_t stream) {
  const float* x  = (const float*)d_in[0];
  const float* wq = (const float*)d_in[1];
  const float* wk = (const float*)d_in[2];
  const float* wv = (const float*)d_in[3];
  float* out = (float*)d_out;
  float* ws  = (float*)d_ws;

  // workspace layout (floats)
  float* wqs  = ws;                  //   512
  float* wks  = ws + 512;            //   512
  float* qs   = ws + 1024;           // 16384
  float* ksS  = ws + 17408;          // 16384 (pre-scaled by SCALE)
  float* kmax = ws + 33792;          //   256
  float* kmin = ws + 34048;          //   256
  float* vbuf = ws + 34304;          // 16384*512 = 8388608 (33.5 MB)

  wsum_kernel <<<  64, 256, 0, stream>>>(wq, wk, wqs, wks);
  qkdot_kernel<<<2048, 256, 0, stream>>>(x, wqs, wks, qs, ksS);
  kstat_kernel<<<  32, 256, 0, stream>>>(ksS, kmax, kmin);
  vproj_kernel<<<4096, 256, 0, stream>>>(x, wv, vbuf);
  attn_kernel <<<2048, 256, 0, stream>>>(qs, ksS, kmax, kmin, vbuf, out);
}